// MP_encoder_50500225466890
// MI455X (gfx1250) — compile-verified
//
#include <hip/hip_runtime.h>
#include <cstddef>

#define HDIM 256
#define BPAD 260                     // LDS row stride (floats): conflict-free, 16B-aligned rows

typedef float v2f __attribute__((ext_vector_type(2)));
typedef float v8f __attribute__((ext_vector_type(8)));

// ---------------------------------------------------------------------------
// Zero-fill (float4 granularity)
// ---------------------------------------------------------------------------
__global__ __launch_bounds__(256) void zero4_kernel(float4* __restrict__ p, long n4) {
    long i = (long)blockIdx.x * blockDim.x + threadIdx.x;
    if (i < n4) p[i] = make_float4(0.f, 0.f, 0.f, 0.f);
}

// ---------------------------------------------------------------------------
// Shared GEMM skeleton: block = 8 waves (256 thr), covers 128 rows x 32 cols.
// W tile (32 rows x 256 K) staged in LDS once per block; each wave computes
// one 16-row m-tile against BOTH 16-col j-tiles (A fragment reused 2x).
// WMMA fragment layouts per CDNA5 ISA:
//   A (f32 16x4): lanes 0-15 hold K=k,k+1 for M=lane; lanes 16-31 K=k+2,k+3.
//   B mirrors it with B[k][j] = W[j][k];  C/D: VGPR r <-> row m0+(lane>=16?8:0)+r.
// ---------------------------------------------------------------------------
__device__ __forceinline__ void stage_W(const float* __restrict__ W, int j0g,
                                        float* __restrict__ Bs) {
    // 32 rows x 64 float4 = 2048 float4; 256 threads -> 8 each.
    for (int t = threadIdx.x; t < 32 * 64; t += 256) {
        int row = t >> 6, c4 = t & 63;
        float4 v = ((const float4*)(W + (size_t)(j0g + row) * HDIM))[c4];
        *(float4*)(Bs + (size_t)row * BPAD + c4 * 4) = v;
    }
}

__global__ __launch_bounds__(256) void gemm_xwt_wmma(const float* __restrict__ A,
                                                     const float* __restrict__ W,
                                                     float* __restrict__ C,
                                                     int nrows) {
    __shared__ __align__(16) float Bs[32 * BPAD];
    const int j0g = (int)blockIdx.x << 5;            // 32 columns per block
    stage_W(W, j0g, Bs);
    __syncthreads();

    const int lane  = threadIdx.x & 31;
    const int wave  = threadIdx.x >> 5;
    const int mtile = blockIdx.y * 8 + wave;
    if (mtile >= (nrows >> 4)) return;               // wave-uniform, after barrier
    const int m0 = mtile << 4;
    const int lr = lane & 15;
    const int kh = (lane >> 4) << 1;                 // 0 or 2

    const float* arow = A + (size_t)(m0 + lr) * HDIM + kh;
    const float* b0p  = Bs + (size_t)lr * BPAD + kh;          // j-tile 0
    const float* b1p  = Bs + (size_t)(16 + lr) * BPAD + kh;   // j-tile 1

    v8f acc0 = {0.f,0.f,0.f,0.f,0.f,0.f,0.f,0.f};
    v8f acc1 = {0.f,0.f,0.f,0.f,0.f,0.f,0.f,0.f};
    for (int k = 0; k < HDIM; k += 4) {
        v2f a  = { arow[k], arow[k + 1] };
        v2f b0 = { b0p[k],  b0p[k + 1]  };
        v2f b1 = { b1p[k],  b1p[k + 1]  };
        acc0 = __builtin_amdgcn_wmma_f32_16x16x4_f32(false, a, false, b0,
                                                     (short)0, acc0, false, false);
        acc1 = __builtin_amdgcn_wmma_f32_16x16x4_f32(false, a, false, b1,
                                                     (short)0, acc1, false, false);
    }

    float* crow = C + (size_t)(m0 + ((lane >> 4) << 3)) * HDIM + j0g + lr;
#pragma unroll
    for (int r = 0; r < 8; ++r) {
        crow[(size_t)r * HDIM]      = acc0[r];
        crow[(size_t)r * HDIM + 16] = acc1[r];
    }
}

// ---------------------------------------------------------------------------
// Fused attention GEMM: colsum[j] += sum_n tanh((E @ fcW^T)[n,j] + fcb[j]).
// Same blocked skeleton; epilogue does tanh + row-reduce + one atomic/lane/tile.
// ---------------------------------------------------------------------------
__global__ __launch_bounds__(256) void fc_tanh_colsum_wmma(const float* __restrict__ E,
                                                           const float* __restrict__ fcW,
                                                           const float* __restrict__ fcb,
                                                           float* __restrict__ colsum,
                                                           int nrows) {
    __shared__ __align__(16) float Bs[32 * BPAD];
    const int j0g = (int)blockIdx.x << 5;
    stage_W(fcW, j0g, Bs);
    __syncthreads();

    const int lane  = threadIdx.x & 31;
    const int wave  = threadIdx.x >> 5;
    const int mtile = blockIdx.y * 8 + wave;
    if (mtile >= (nrows >> 4)) return;
    const int m0 = mtile << 4;
    const int lr = lane & 15;
    const int kh = (lane >> 4) << 1;

    const float* arow = E + (size_t)(m0 + lr) * HDIM + kh;
    const float* b0p  = Bs + (size_t)lr * BPAD + kh;
    const float* b1p  = Bs + (size_t)(16 + lr) * BPAD + kh;

    v8f acc0 = {0.f,0.f,0.f,0.f,0.f,0.f,0.f,0.f};
    v8f acc1 = {0.f,0.f,0.f,0.f,0.f,0.f,0.f,0.f};
    for (int k = 0; k < HDIM; k += 4) {
        v2f a  = { arow[k], arow[k + 1] };
        v2f b0 = { b0p[k],  b0p[k + 1]  };
        v2f b1 = { b1p[k],  b1p[k + 1]  };
        acc0 = __builtin_amdgcn_wmma_f32_16x16x4_f32(false, a, false, b0,
                                                     (short)0, acc0, false, false);
        acc1 = __builtin_amdgcn_wmma_f32_16x16x4_f32(false, a, false, b1,
                                                     (short)0, acc1, false, false);
    }

    float bias0 = fcb[j0g + lr];
    float bias1 = fcb[j0g + 16 + lr];
    float s0 = 0.f, s1 = 0.f;
#pragma unroll
    for (int r = 0; r < 8; ++r) {
        s0 += tanhf(acc0[r] + bias0);
        s1 += tanhf(acc1[r] + bias1);
    }
    atomicAdd(&colsum[j0g + lr], s0);
    atomicAdd(&colsum[j0g + 16 + lr], s1);
}

// ---------------------------------------------------------------------------
// SpMM: out[dst] += val[e] * fts[src].  64 threads/edge, float4 per thread.
// Atomics land in L2 (each output embed is 51.2MB < 192MB L2).
// ---------------------------------------------------------------------------
__global__ __launch_bounds__(256) void spmm_kernel(const int* __restrict__ idx,
                                                   const float* __restrict__ val,
                                                   const float* __restrict__ fts,
                                                   float* __restrict__ out,
                                                   int nedges) {
    int t = blockIdx.x * blockDim.x + threadIdx.x;
    int e = t >> 6;
    if (e >= nedges) return;
    int q   = t & 63;
    int dst = idx[e];                 // idx[0][e]
    int src = idx[nedges + e];        // idx[1][e]
    float v = val[e];
    float4 f = ((const float4*)(fts + (size_t)src * HDIM))[q];
    float* o = out + (size_t)dst * HDIM + q * 4;
    atomicAdd(o + 0, v * f.x);
    atomicAdd(o + 1, v * f.y);
    atomicAdd(o + 2, v * f.z);
    atomicAdd(o + 3, v * f.w);
}

// ---------------------------------------------------------------------------
// In-place bias add + PReLU (shared scalar slope).
// ---------------------------------------------------------------------------
__global__ __launch_bounds__(256) void bias_prelu_kernel(float4* __restrict__ e,
                                                         const float4* __restrict__ b4,
                                                         const float* __restrict__ a,
                                                         long n4) {
    long i = (long)blockIdx.x * blockDim.x + threadIdx.x;
    if (i >= n4) return;
    float s  = a[0];
    float4 v = e[i];
    float4 b = b4[i & 63];            // HDIM/4 == 64 float4 per row
    v.x += b.x; v.y += b.y; v.z += b.z; v.w += b.w;
    v.x = v.x > 0.f ? v.x : s * v.x;
    v.y = v.y > 0.f ? v.y : s * v.y;
    v.z = v.z > 0.f ? v.z : s * v.z;
    v.w = v.w > 0.f ? v.w : s * v.w;
    e[i] = v;
}

// ---------------------------------------------------------------------------
// Per-side softmax over the two metapath scores.
// blockIdx.x = side (0: d, 1: p), 256 threads.
// ---------------------------------------------------------------------------
__global__ __launch_bounds__(HDIM) void score_softmax_kernel(const float* __restrict__ colsum,
                                                             const float* __restrict__ attd,
                                                             const float* __restrict__ attp,
                                                             float* __restrict__ beta,
                                                             float invN) {
    __shared__ float r0[HDIM], r1[HDIM];
    int side = blockIdx.x;
    int j = threadIdx.x;
    const float* att = (side == 0) ? attd : attp;
    const float* c0 = colsum + (size_t)(side * 2 + 0) * HDIM;
    const float* c1 = colsum + (size_t)(side * 2 + 1) * HDIM;
    r0[j] = c0[j] * invN * att[j];
    r1[j] = c1[j] * invN * att[j];
    __syncthreads();
    for (int st = HDIM / 2; st > 0; st >>= 1) {
        if (j < st) { r0[j] += r0[j + st]; r1[j] += r1[j + st]; }
        __syncthreads();
    }
    if (j == 0) {
        float s0 = r0[0], s1 = r1[0];
        float m = fmaxf(s0, s1);
        float e0 = expf(s0 - m), e1 = expf(s1 - m);
        float inv = 1.f / (e0 + e1);
        beta[side * 2 + 0] = e0 * inv;
        beta[side * 2 + 1] = e1 * inv;
    }
}

// ---------------------------------------------------------------------------
// z = beta[0]*e0 + beta[1]*e1
// ---------------------------------------------------------------------------
__global__ __launch_bounds__(256) void combine_kernel(const float4* __restrict__ e0,
                                                      const float4* __restrict__ e1,
                                                      const float* __restrict__ beta,
                                                      float4* __restrict__ z,
                                                      long n4) {
    long i = (long)blockIdx.x * blockDim.x + threadIdx.x;
    if (i >= n4) return;
    float b0 = beta[0], b1 = beta[1];
    float4 a = e0[i], b = e1[i], o;
    o.x = b0 * a.x + b1 * b.x;
    o.y = b0 * a.y + b1 * b.y;
    o.z = b0 * a.z + b1 * b.z;
    o.w = b0 * a.w + b1 * b.w;
    z[i] = o;
}

// ---------------------------------------------------------------------------
// Host-side orchestration
// ---------------------------------------------------------------------------
extern "C" void kernel_launch(void* const* d_in, const int* in_sizes, int n_in,
                              void* d_out, int out_size, void* d_ws, size_t ws_size,
                              hipStream_t stream) {
    (void)n_in; (void)out_size; (void)ws_size;

    const int N = in_sizes[0] / HDIM;      // 50000
    const int E = in_sizes[3];             // 800000
    const long NH = (long)N * HDIM;        // 12.8M

    const float* h_d = (const float*)d_in[0];
    const float* h_p = (const float*)d_in[1];
    const int*   idx[4] = { (const int*)d_in[2], (const int*)d_in[4],
                            (const int*)d_in[6], (const int*)d_in[8] };
    const float* val[4] = { (const float*)d_in[3], (const float*)d_in[5],
                            (const float*)d_in[7], (const float*)d_in[9] };
    const float* Wm[4]  = { (const float*)d_in[10], (const float*)d_in[13],
                            (const float*)d_in[16], (const float*)d_in[19] };
    const float* bm[4]  = { (const float*)d_in[11], (const float*)d_in[14],
                            (const float*)d_in[17], (const float*)d_in[20] };
    const float* am[4]  = { (const float*)d_in[12], (const float*)d_in[15],
                            (const float*)d_in[18], (const float*)d_in[21] };
    const float* fcW[2] = { (const float*)d_in[22], (const float*)d_in[25] };
    const float* fcb[2] = { (const float*)d_in[23], (const float*)d_in[26] };
    const float* att[2] = { (const float*)d_in[24], (const float*)d_in[27] };

    // Workspace layout (floats): [embed0..embed3 | colsum(4*H) | beta(4) | fts]
    float* ws      = (float*)d_ws;
    float* embed[4];
    for (int i = 0; i < 4; ++i) embed[i] = ws + (long)i * NH;
    float* colsum  = ws + 4 * NH;
    float* beta    = colsum + 4 * HDIM;
    float* fts     = beta + 4;             // byte offset 16-aligned

    float* z_d = (float*)d_out;
    float* z_p = (float*)d_out + NH;

    // 1. Zero embeds + colsum + beta (contiguous region).
    {
        long n4 = NH + (4 * HDIM + 4) / 4;     // 4*NH floats + 1028 floats, in float4
        int blocks = (int)((n4 + 255) / 256);
        zero4_kernel<<<blocks, 256, 0, stream>>>((float4*)ws, n4);
    }

    const int mtiles = N >> 4;                       // 3125
    dim3 ggrid(HDIM / 32, (mtiles + 7) / 8);         // (8, 391)

    // 2. Per metapath: GEMM (h @ W^T) then SpMM scatter-add.
    for (int mp = 0; mp < 4; ++mp) {
        const float* h = (mp < 2) ? h_d : h_p;
        gemm_xwt_wmma<<<ggrid, 256, 0, stream>>>(h, Wm[mp], fts, N);
        int sblocks = (int)(((long)E * 64 + 255) / 256);
        spmm_kernel<<<sblocks, 256, 0, stream>>>(idx[mp], val[mp], fts, embed[mp], E);
    }

    // 3. Bias + PReLU per metapath.
    {
        long n4 = NH / 4;
        int blocks = (int)((n4 + 255) / 256);
        for (int mp = 0; mp < 4; ++mp)
            bias_prelu_kernel<<<blocks, 256, 0, stream>>>((float4*)embed[mp],
                                                          (const float4*)bm[mp],
                                                          am[mp], n4);
    }

    // 4. Fused attention GEMM + tanh + column-sum per (side, path).
    for (int side = 0; side < 2; ++side)
        for (int p = 0; p < 2; ++p)
            fc_tanh_colsum_wmma<<<ggrid, 256, 0, stream>>>(embed[side * 2 + p],
                                                           fcW[side], fcb[side],
                                                           colsum + (long)(side * 2 + p) * HDIM,
                                                           N);

    // 5. Per-side softmax over metapath scores.
    score_softmax_kernel<<<2, HDIM, 0, stream>>>(colsum, att[0], att[1], beta, 1.0f / (float)N);

    // 6. Weighted combine into outputs.
    {
        long n4 = NH / 4;
        int blocks = (int)((n4 + 255) / 256);
        combine_kernel<<<blocks, 256, 0, stream>>>((const float4*)embed[0],
                                                   (const float4*)embed[1],
                                                   beta + 0, (float4*)z_d, n4);
        combine_kernel<<<blocks, 256, 0, stream>>>((const float4*)embed[2],
                                                   (const float4*)embed[3],
                                                   beta + 2, (float4*)z_p, n4);
    }
}